// ModelNew_4810363372240
// MI455X (gfx1250) — compile-verified
//
#include <hip/hip_runtime.h>
#include <stdint.h>

#define ROW_LEN 512

// Masked Hillis-Steele scan, one 512-thread block (16 wave32) per row.
// CDNA5 path: the row of x is streamed global -> LDS by the async copy
// engine (b128 granularity, ASYNCcnt-tracked, no VGPR round trip) while the
// int32 mask rides the normal vmem path with an NT hint. The 9 effective
// strides (s = 512 is a no-op since tid < 512) run out of a ping-pong LDS
// buffer with exactly one barrier per stride and branch-free cndmask gating.
__global__ __launch_bounds__(ROW_LEN) void ModelNew_masked_hs_kernel(
    const float* __restrict__ x,
    const int* __restrict__ mask,
    float* __restrict__ out) {
  __shared__ __align__(16) float buf[2][ROW_LEN];

  const int tid = threadIdx.x;
  const size_t base = (size_t)blockIdx.x * ROW_LEN + tid;

  // --- CDNA5 async global -> LDS copy of the row --------------------------
  // Waves 0..3 (tid < 128, wave-uniform) each issue one b128 transfer per
  // lane: 512 B per wave request, 4 requests per block for the 2 KB row.
  // the vdst VGPR holds the LDS byte address; the low 32 bits of a generic pointer to
  // an LDS object are the workgroup-relative LDS byte offset on gfx1250.
  if (tid < (ROW_LEN / 4)) {
    unsigned lds_addr = (unsigned)(uintptr_t)(&buf[0][tid * 4]);
    const float* gaddr = x + (size_t)blockIdx.x * ROW_LEN + tid * 4;
    asm volatile("global_load_async_to_lds_b128 %0, %1, off"
                 :
                 : "v"(lds_addr), "v"(gaddr)
                 : "memory");
  }

  // Mask load overlaps the async copy. Streaming input: non-temporal.
  const int mraw = __builtin_nontemporal_load(mask + base);
  const bool m = (mraw != 0);

  // Per-wave async completion, then a block barrier so every wave sees the
  // LDS row written by waves 0..3.
  asm volatile("s_wait_asynccnt 0" ::: "memory");
  __syncthreads();

  // Apply the mask to our own slot: unconditional ds_load + v_cndmask (no
  // exec-mask branching), self-read/self-write, then publish.
  float xv = buf[0][tid];
  float v = m ? xv : 0.0f;
  buf[0][tid] = v;
  __syncthreads();

  // --- 9 strides, ping-pong buffers, one barrier per stride ---------------
  // Masked-out lanes never accumulate (stay 0), which makes the reference's
  // mask[i-s] test redundant. Stride reads use a wrapped index so the
  // ds_load is unconditional; the discard happens in the cndmask.
  int src = 0;
#pragma unroll
  for (int s = 1; s < ROW_LEN; s <<= 1) {
    float prev = buf[src][(tid - s) & (ROW_LEN - 1)];
    bool take = m && (tid >= s);
    v += take ? prev : 0.0f;
    buf[src ^ 1][tid] = v;
    __syncthreads();
    src ^= 1;
  }

  // Streaming output: non-temporal store (256 MB stream >> 192 MB L2).
  __builtin_nontemporal_store(v, out + base);
}

extern "C" void kernel_launch(void* const* d_in, const int* in_sizes, int n_in,
                              void* d_out, int out_size, void* d_ws, size_t ws_size,
                              hipStream_t stream) {
  (void)n_in; (void)d_ws; (void)ws_size; (void)out_size;

  const float* x = (const float*)d_in[0];
  const int* mask = (const int*)d_in[1];
  float* out = (float*)d_out;

  const int n_rows = in_sizes[0] / ROW_LEN;  // B = 131072
  if (n_rows <= 0) return;

  ModelNew_masked_hs_kernel<<<dim3((unsigned)n_rows), dim3(ROW_LEN), 0, stream>>>(
      x, mask, out);
}